// EnhancedGAT_22497038696816
// MI455X (gfx1250) — compile-verified
//
#include <hip/hip_runtime.h>
#include <hip/hip_bf16.h>
#include <math.h>

#define N_NODES 20000
#define N_EDGES 640000
#define F_IN    128
#define HID     64
#define HEADS   4
#define NGRAPH  256
#define OUT_F   32
#define F1      (HEADS*HID)   // 256

typedef __attribute__((ext_vector_type(2))) float v2f;
typedef __attribute__((ext_vector_type(8))) float v8f;

// ---- monotone float <-> u32 key for atomicMax-based segment max ----
__device__ __forceinline__ unsigned fkey(float f) {
    unsigned u = __float_as_uint(f);
    return (u & 0x80000000u) ? ~u : (u | 0x80000000u);
}
__device__ __forceinline__ float finv(unsigned k) {
    unsigned u = (k & 0x80000000u) ? (k & 0x7FFFFFFFu) : ~k;
    return __uint_as_float(u);
}

// ---- generic u32 fill ----
__global__ void fill_u32(unsigned* __restrict__ p, unsigned v, long n) {
    long t = (long)blockIdx.x * blockDim.x + threadIdx.x;
    if (t < n) p[t] = v;
}

// ---- fp32 WMMA GEMM: C[M,N] = A[M,K] * B[K,N], row-major, 16x16 tile/wave ----
__global__ void wmma_gemm_f32(const float* __restrict__ A, const float* __restrict__ B,
                              float* __restrict__ C, int M, int N, int K) {
    int lane = threadIdx.x & 31;
    int wid  = blockIdx.x * (blockDim.x >> 5) + (threadIdx.x >> 5);  // wave-uniform
    int tilesN = N >> 4;
    int totalT = (M >> 4) * tilesN;
    if (wid >= totalT) return;                                       // wave-uniform branch
    int tm = (wid / tilesN) << 4;
    int tn = (wid % tilesN) << 4;
    int r  = lane & 15;          // row (A) / col (B,C)
    int hi = lane >> 4;          // k-half selector
    const float* ap = A + (long)(tm + r) * K;
    const float* bp = B + tn + r;
    v8f acc = {};
    for (int k0 = 0; k0 < K; k0 += 4) {
        v2f a, b;
        a.x = ap[k0 + 2*hi];
        a.y = ap[k0 + 2*hi + 1];
        b.x = bp[(long)(k0 + 2*hi) * N];
        b.y = bp[(long)(k0 + 2*hi + 1) * N];
        acc = __builtin_amdgcn_wmma_f32_16x16x4_f32(false, a, false, b,
                                                    (short)0, acc, false, false);
    }
    float* cp = C + (long)(tm + hi*8) * N + tn + r;
#pragma unroll
    for (int i = 0; i < 8; ++i) cp[(long)i * N] = acc[i];
}

// ---- per-node attention logits: a_src[n,h], a_dst[n,h] ----
__global__ void node_att(const float* __restrict__ h, const float* __restrict__ att_s,
                         const float* __restrict__ att_d, float* __restrict__ a_src,
                         float* __restrict__ a_dst, int n, int heads, int ch) {
    int t = blockIdx.x * blockDim.x + threadIdx.x;
    if (t >= n * heads) return;
    int node = t / heads, hd = t - node * heads;
    const float* hp = h + (long)node * heads * ch + hd * ch;
    const float* as = att_s + hd * ch;
    const float* ad = att_d + hd * ch;
    float s = 0.f, d = 0.f;
    for (int c = 0; c < ch; ++c) { float v = hp[c]; s += v * as[c]; d += v * ad[c]; }
    a_src[t] = s; a_dst[t] = d;
}

// ---- edge-attr rank-1 coefficients: ce[h] = dot(We_h, att_edge_h) ----
__global__ void compute_ce(const float* __restrict__ We1, const float* __restrict__ ae1,
                           const float* __restrict__ We2, const float* __restrict__ ae2,
                           float* __restrict__ ce) {
    int t = threadIdx.x;
    if (t < HEADS) {
        float s = 0.f;
        for (int c = 0; c < HID; ++c) s += We1[t*HID + c] * ae1[t*HID + c];
        ce[t] = s;
    } else if (t == HEADS) {
        float s = 0.f;
        for (int c = 0; c < HID; ++c) s += We2[c] * ae2[c];
        ce[HEADS] = s;
    }
}

// ---- pass A: per-edge logits + segment max (monotone-key atomicMax) ----
__global__ void edge_logits(const int* __restrict__ src, const int* __restrict__ dst,
                            const float* __restrict__ eattr, const float* __restrict__ a_src,
                            const float* __restrict__ a_dst, const float* __restrict__ ce,
                            float* __restrict__ alpha, unsigned* __restrict__ mkey,
                            int E, int heads) {
    int t = blockIdx.x * blockDim.x + threadIdx.x;
    if (t >= E * heads) return;
    int e = t / heads, hd = t - e * heads;
    int s = src[e], d = dst[e];
    float a = a_src[s*heads + hd] + a_dst[d*heads + hd] + eattr[e] * ce[hd];
    a = (a > 0.f) ? a : 0.2f * a;                   // leaky_relu(0.2)
    alpha[t] = a;
    atomicMax(&mkey[d*heads + hd], fkey(a));
}

// ---- pass B: p = exp(a - m[dst]); denom[dst] += p ----
__global__ void edge_exp(const int* __restrict__ dst, float* __restrict__ alpha,
                         const unsigned* __restrict__ mkey, float* __restrict__ denom,
                         int E, int heads) {
    int t = blockIdx.x * blockDim.x + threadIdx.x;
    if (t >= E * heads) return;
    int e = t / heads, hd = t - e * heads;
    int d = dst[e];
    float p = expf(alpha[t] - finv(mkey[d*heads + hd]));
    alpha[t] = p;
    atomicAdd(&denom[d*heads + hd], p);
}

// ---- pass C: out[dst] += h[src] * (p / denom[dst]); float4 per thread ----
__global__ void edge_aggregate(const int* __restrict__ src, const int* __restrict__ dst,
                               const float* __restrict__ h, const float* __restrict__ alpha,
                               const float* __restrict__ denom, float* __restrict__ out,
                               int E, int heads, int ch) {
    int F = heads * ch;
    int q = F >> 2;                                  // float4 groups per edge
    long t = (long)blockIdx.x * blockDim.x + threadIdx.x;
    if (t >= (long)E * q) return;
    int e = (int)(t / q);
    int j = (int)(t - (long)e * q);
    int hd = (j << 2) / ch;
    int s = src[e], d = dst[e];
    float w = alpha[(long)e*heads + hd] / (denom[d*heads + hd] + 1e-16f);
    float4 v = ((const float4*)(h + (long)s * F))[j];
    float* op = out + (long)d * F + (j << 2);
    atomicAdd(op + 0, v.x * w);
    atomicAdd(op + 1, v.y * w);
    atomicAdd(op + 2, v.z * w);
    atomicAdd(op + 3, v.w * w);
}

// ---- x = relu(x + b), in place ----
__global__ void bias_relu(float* __restrict__ x, const float* __restrict__ b, long n, int F) {
    long t = (long)blockIdx.x * blockDim.x + threadIdx.x;
    if (t >= n) return;
    int k = (int)(t % F);
    float v = x[t] + b[k];
    x[t] = (v > 0.f) ? v : 0.f;
}

// ---- global mean pool (sum + count), float4 per thread ----
__global__ void pool_sum(const float* __restrict__ h, const int* __restrict__ batch,
                         float* __restrict__ sums, float* __restrict__ cnt, int n) {
    int t = blockIdx.x * blockDim.x + threadIdx.x;
    if (t >= n * (HID >> 2)) return;
    int node = t >> 4, j = t & 15;                  // HID/4 == 16
    int g = batch[node];
    float4 v = ((const float4*)(h + (long)node * HID))[j];
    float* sp = sums + (long)g * HID + (j << 2);
    atomicAdd(sp + 0, v.x);
    atomicAdd(sp + 1, v.y);
    atomicAdd(sp + 2, v.z);
    atomicAdd(sp + 3, v.w);
    if (j == 0) atomicAdd(&cnt[g], 1.0f);
}

// ---- final FC: out[g,o] = (sums[g]/max(cnt,1)) . fcW[:,o] + fcb[o] ----
__global__ void final_fc(const float* __restrict__ sums, const float* __restrict__ cnt,
                         const float* __restrict__ fcW, const float* __restrict__ fcb,
                         float* __restrict__ out) {
    int t = blockIdx.x * blockDim.x + threadIdx.x;
    if (t >= NGRAPH * OUT_F) return;
    int g = t / OUT_F, o = t - g * OUT_F;
    float c = cnt[g]; c = (c > 1.f) ? c : 1.f;
    float inv = 1.f / c;
    float acc = fcb[o];
    for (int k = 0; k < HID; ++k) acc += sums[(long)g*HID + k] * inv * fcW[k*OUT_F + o];
    out[t] = acc;
}

extern "C" void kernel_launch(void* const* d_in, const int* in_sizes, int n_in,
                              void* d_out, int out_size, void* d_ws, size_t ws_size,
                              hipStream_t stream) {
    const float* x      = (const float*)d_in[0];
    const int*   eidx   = (const int*)  d_in[1];
    const float* eattr  = (const float*)d_in[2];
    const int*   batch  = (const int*)  d_in[3];
    const float* W1     = (const float*)d_in[4];
    const float* as1    = (const float*)d_in[5];
    const float* ad1    = (const float*)d_in[6];
    const float* We1    = (const float*)d_in[7];
    const float* ae1    = (const float*)d_in[8];
    const float* b1     = (const float*)d_in[9];
    const float* W2     = (const float*)d_in[10];
    const float* as2    = (const float*)d_in[11];
    const float* ad2    = (const float*)d_in[12];
    const float* We2    = (const float*)d_in[13];
    const float* ae2    = (const float*)d_in[14];
    const float* b2     = (const float*)d_in[15];
    const float* fcW    = (const float*)d_in[16];
    const float* fcb    = (const float*)d_in[17];
    float* out = (float*)d_out;

    const int* src = eidx;            // edge_index[0,:]
    const int* dst = eidx + N_EDGES;  // edge_index[1,:]

    // ---- workspace carve-out (256B aligned) ----
    char* ws = (char*)d_ws;
    size_t off = 0;
    auto carve = [&](size_t bytes) { size_t o = off; off = (off + bytes + 255) & ~(size_t)255; return o; };
    float*    h1     = (float*)   (ws + carve((size_t)N_NODES * F1  * 4));
    float*    out1   = (float*)   (ws + carve((size_t)N_NODES * F1  * 4));
    float*    h2     = (float*)   (ws + carve((size_t)N_NODES * HID * 4));
    float*    out2   = (float*)   (ws + carve((size_t)N_NODES * HID * 4));
    float*    asrc1  = (float*)   (ws + carve((size_t)N_NODES * HEADS * 4));
    float*    adst1  = (float*)   (ws + carve((size_t)N_NODES * HEADS * 4));
    unsigned* mkey1  = (unsigned*)(ws + carve((size_t)N_NODES * HEADS * 4));
    float*    den1   = (float*)   (ws + carve((size_t)N_NODES * HEADS * 4));
    float*    alpha1 = (float*)   (ws + carve((size_t)N_EDGES * HEADS * 4));
    float*    asrc2  = (float*)   (ws + carve((size_t)N_NODES * 4));
    float*    adst2  = (float*)   (ws + carve((size_t)N_NODES * 4));
    unsigned* mkey2  = (unsigned*)(ws + carve((size_t)N_NODES * 4));
    float*    den2   = (float*)   (ws + carve((size_t)N_NODES * 4));
    float*    alpha2 = (float*)   (ws + carve((size_t)N_EDGES * 4));
    float*    ce     = (float*)   (ws + carve(8 * 4));
    float*    sums   = (float*)   (ws + carve((size_t)NGRAPH * HID * 4));
    float*    cnt    = (float*)   (ws + carve((size_t)NGRAPH * 4));
    (void)ws_size; (void)n_in; (void)in_sizes; (void)out_size;

    // key(-inf): bits(-inf)=0xFF800000 (negative) -> ~bits = 0x007FFFFF
    const unsigned KNEG = 0x007FFFFFu;

    auto blk = [](long n, int b) { return (unsigned)((n + b - 1) / b); };

    // ---- init accumulators ----
    fill_u32<<<blk((long)N_NODES*HEADS,256),256,0,stream>>>(mkey1, KNEG, (long)N_NODES*HEADS);
    fill_u32<<<blk((long)N_NODES*HEADS,256),256,0,stream>>>((unsigned*)den1, 0u, (long)N_NODES*HEADS);
    fill_u32<<<blk((long)N_NODES*F1,256),256,0,stream>>>((unsigned*)out1, 0u, (long)N_NODES*F1);
    fill_u32<<<blk(N_NODES,256),256,0,stream>>>(mkey2, KNEG, N_NODES);
    fill_u32<<<blk(N_NODES,256),256,0,stream>>>((unsigned*)den2, 0u, N_NODES);
    fill_u32<<<blk((long)N_NODES*HID,256),256,0,stream>>>((unsigned*)out2, 0u, (long)N_NODES*HID);
    fill_u32<<<blk(NGRAPH*HID,256),256,0,stream>>>((unsigned*)sums, 0u, NGRAPH*HID);
    fill_u32<<<blk(NGRAPH,256),256,0,stream>>>((unsigned*)cnt, 0u, NGRAPH);

    // ---- layer 1 ----
    wmma_gemm_f32<<<(N_NODES/16)*(F1/16)/8, 256, 0, stream>>>(x, W1, h1, N_NODES, F1, F_IN);
    node_att<<<blk((long)N_NODES*HEADS,256),256,0,stream>>>(h1, as1, ad1, asrc1, adst1,
                                                            N_NODES, HEADS, HID);
    compute_ce<<<1, 32, 0, stream>>>(We1, ae1, We2, ae2, ce);
    edge_logits<<<blk((long)N_EDGES*HEADS,256),256,0,stream>>>(src, dst, eattr, asrc1, adst1,
                                                               ce, alpha1, mkey1, N_EDGES, HEADS);
    edge_exp<<<blk((long)N_EDGES*HEADS,256),256,0,stream>>>(dst, alpha1, mkey1, den1,
                                                            N_EDGES, HEADS);
    edge_aggregate<<<blk((long)N_EDGES*(F1/4),256),256,0,stream>>>(src, dst, h1, alpha1, den1,
                                                                   out1, N_EDGES, HEADS, HID);
    bias_relu<<<blk((long)N_NODES*F1,256),256,0,stream>>>(out1, b1, (long)N_NODES*F1, F1);

    // ---- layer 2 (heads=1) ----
    wmma_gemm_f32<<<(N_NODES/16)*(HID/16)/8, 256, 0, stream>>>(out1, W2, h2, N_NODES, HID, F1);
    node_att<<<blk(N_NODES,256),256,0,stream>>>(h2, as2, ad2, asrc2, adst2, N_NODES, 1, HID);
    edge_logits<<<blk(N_EDGES,256),256,0,stream>>>(src, dst, eattr, asrc2, adst2,
                                                   ce + HEADS, alpha2, mkey2, N_EDGES, 1);
    edge_exp<<<blk(N_EDGES,256),256,0,stream>>>(dst, alpha2, mkey2, den2, N_EDGES, 1);
    edge_aggregate<<<blk((long)N_EDGES*(HID/4),256),256,0,stream>>>(src, dst, h2, alpha2, den2,
                                                                    out2, N_EDGES, 1, HID);
    bias_relu<<<blk((long)N_NODES*HID,256),256,0,stream>>>(out2, b2, (long)N_NODES*HID, HID);

    // ---- pool + FC ----
    pool_sum<<<blk((long)N_NODES*(HID/4),256),256,0,stream>>>(out2, batch, sums, cnt, N_NODES);
    final_fc<<<blk(NGRAPH*OUT_F,256),256,0,stream>>>(sums, cnt, fcW, fcb, out);
}